// BoW_11527692222508
// MI455X (gfx1250) — compile-verified
//
#include <hip/hip_runtime.h>

#define TOTAL_TOKENS 819200
#define NUM_SENT     16384
#define DIM          128

typedef __attribute__((ext_vector_type(2))) float v2f;
typedef __attribute__((ext_vector_type(8))) float v8f;

// ---------------------------------------------------------------------------
// Kernel 1: fused embedding gather + ragged (sorted-run) segment sum.
// One 128-thread block per sentence; thread t owns dim t. segment_ids is
// sorted, so segment s occupies [lower_bound(s), lower_bound(s+1)).
// Embedding row loads are fully coalesced (128 lanes x 4B = 512B).
// ---------------------------------------------------------------------------
__global__ __launch_bounds__(128) void seg_pool_kernel(
    const int*   __restrict__ tokens,
    const int*   __restrict__ seg,
    const float* __restrict__ emb,
    float*       __restrict__ bow)
{
    __shared__ int bounds[2];
    const int s   = blockIdx.x;
    const int tid = threadIdx.x;

    if (tid < 2) {
        const int key = s + tid;
        int lo = 0, hi = TOTAL_TOKENS;
        while (lo < hi) {
            const int mid = (lo + hi) >> 1;
            const int v   = seg[mid];
            if (v < key) lo = mid + 1; else hi = mid;
        }
        bounds[tid] = lo;
    }
    __syncthreads();

    const int start = bounds[0];
    const int end   = bounds[1];

    float acc = 0.0f;
    int i = start;
    // Unroll x4: issue 4 independent token + row loads to hide gather latency.
    for (; i + 4 <= end; i += 4) {
        const int t0 = tokens[i + 0];
        const int t1 = tokens[i + 1];
        const int t2 = tokens[i + 2];
        const int t3 = tokens[i + 3];
        const float e0 = emb[t0 * DIM + tid];
        const float e1 = emb[t1 * DIM + tid];
        const float e2 = emb[t2 * DIM + tid];
        const float e3 = emb[t3 * DIM + tid];
        acc += e0 + e1 + e2 + e3;
    }
    for (; i < end; ++i) {
        acc += emb[tokens[i] * DIM + tid];
    }
    bow[s * DIM + tid] = acc;
}

// ---------------------------------------------------------------------------
// Kernel 2: relu(bow @ W + b) with full-f32 WMMA (V_WMMA_F32_16X16X4_F32).
// One block per 16-row M-tile of bow; the 16x128 A tile is staged in LDS and
// shared by 8 waves, each computing one 16x16 output tile (N tile = wave id)
// via 32 chained K=4 WMMAs. W (64 KB) is L2-resident.
//
// Fragment layouts (ISA 7.12.2, 32-bit):
//   A 16x4 : lanes 0-15 -> M=lane,   VGPR{0,1}=K{0,1}; lanes 16-31 -> K{2,3}
//   B 4x16 : lanes 0-15 -> N=lane,   VGPR{0,1}=K{0,1}; lanes 16-31 -> K{2,3}
//   C/D    : VGPR r -> row r (lanes 0-15) / row r+8 (lanes 16-31), N=lane&15
// ---------------------------------------------------------------------------
__global__ __launch_bounds__(256) void gemm_bias_relu_kernel(
    const float* __restrict__ bow,
    const float* __restrict__ W,
    const float* __restrict__ bias,
    float*       __restrict__ out)
{
    __shared__ float Atile[16 * DIM];  // 8 KB

    const int mtile = blockIdx.x;
    const int tid   = threadIdx.x;

    // Cooperative load of the 16x128 A tile (2048 floats / 256 threads).
    #pragma unroll
    for (int j = 0; j < 8; ++j) {
        const int idx = tid + j * 256;
        Atile[idx] = bow[mtile * 16 * DIM + idx];
    }
    __syncthreads();

    const int wave  = tid >> 5;        // 0..7 -> N tile
    const int lane  = tid & 31;
    const int hi    = lane >> 4;       // 0: K lanes {0,1}; 1: K lanes {2,3}
    const int m     = lane & 15;       // row within A tile / col within N tile
    const int n     = wave * 16 + m;   // global output column
    const int kk    = hi * 2;

    v8f c = {};
    #pragma unroll 8
    for (int k = 0; k < DIM; k += 4) {
        v2f a, bf;
        a.x  = Atile[m * DIM + (k + kk)];
        a.y  = Atile[m * DIM + (k + kk + 1)];
        bf.x = W[(k + kk)     * DIM + n];
        bf.y = W[(k + kk + 1) * DIM + n];
        // 8 args: (neg_a, A, neg_b, B, c_mod, C, reuse_a, reuse_b)
        c = __builtin_amdgcn_wmma_f32_16x16x4_f32(
                false, a, false, bf, (short)0, c, false, false);
    }

    const float bv      = bias[n];
    const int   rowbase = mtile * 16 + hi * 8;
    #pragma unroll
    for (int r = 0; r < 8; ++r) {
        const float v = c[r] + bv;
        out[(rowbase + r) * DIM + n] = v > 0.0f ? v : 0.0f;
    }
}

// ---------------------------------------------------------------------------
// Launch: inputs in setup_inputs() order:
//   d_in[0]=tokens(i32), d_in[1]=segment_ids(i32), d_in[2]=embedding_table(f32),
//   d_in[3]=W(f32), d_in[4]=b(f32).  d_out = [16384,128] f32.
//   d_ws: 16384*128*4 = 8 MB intermediate "bow".
// ---------------------------------------------------------------------------
extern "C" void kernel_launch(void* const* d_in, const int* in_sizes, int n_in,
                              void* d_out, int out_size, void* d_ws, size_t ws_size,
                              hipStream_t stream) {
    const int*   tokens = (const int*)d_in[0];
    const int*   seg    = (const int*)d_in[1];
    const float* emb    = (const float*)d_in[2];
    const float* W      = (const float*)d_in[3];
    const float* b      = (const float*)d_in[4];
    float*       out    = (float*)d_out;
    float*       bow    = (float*)d_ws;

    seg_pool_kernel<<<NUM_SENT, 128, 0, stream>>>(tokens, seg, emb, bow);
    gemm_bias_relu_kernel<<<NUM_SENT / 16, 256, 0, stream>>>(bow, W, b, out);
}